// CliqueFluxNet_17360257810476
// MI455X (gfx1250) — compile-verified
//
#include <hip/hip_runtime.h>
#include <math.h>

typedef __attribute__((ext_vector_type(2))) float v2f;
typedef __attribute__((ext_vector_type(8))) float v8f;

#define IN_DIM 512
#define HID    256

// ---------------------------------------------------------------------------
// Degree / normalization
// ---------------------------------------------------------------------------
__global__ void deg_init_kernel(float* __restrict__ deg, int n) {
    int i = blockIdx.x * blockDim.x + threadIdx.x;
    if (i < n) deg[i] = 1.0f;          // self-loop contribution
}

__global__ void deg_accum_kernel(const long long* __restrict__ dst,
                                 float* __restrict__ deg, int e) {
    int i = blockIdx.x * blockDim.x + threadIdx.x;
    if (i < e) atomicAdd(&deg[(int)dst[i]], 1.0f);
}

__global__ void deg_to_dinv_kernel(float* __restrict__ deg, int n) {
    int i = blockIdx.x * blockDim.x + threadIdx.x;
    if (i < n) {
        float d = deg[i];
        deg[i] = (d > 0.0f) ? rsqrtf(d) : 0.0f;
    }
}

// ---------------------------------------------------------------------------
// fp32 WMMA GEMM: H[n][HID] = X[n][K] @ W[K][HID]
// One wave -> 32(M) x 64(N) tile: 2 A fragments x 4 B fragments ->
// 8 v8f accumulators (64 VGPRs), 8 WMMAs per K-step of 4 (no spills).
// A layout (16x4 f32): lane<16 -> {K=k0,k0+1}, lane>=16 -> {K=k0+2,k0+3}
// B layout (4x16 f32): same half/K split, column = lane & 15
// C/D layout: element (M = r + 8*half, N = lane & 15) lives in VGPR r.
// ---------------------------------------------------------------------------
__global__ void gemm_wmma_f32_kernel(const float* __restrict__ X,
                                     const float* __restrict__ W,
                                     float* __restrict__ H,
                                     int nrows, int K) {
    const int tilesN = HID / 64;                      // 4 strips of 64 cols
    const int tilesM = (nrows + 31) >> 5;             // 32 rows per wave tile
    const int wavesPerBlock = blockDim.x >> 5;
    int waveId = blockIdx.x * wavesPerBlock + (threadIdx.x >> 5);
    if (waveId >= tilesM * tilesN) return;

    int tm = waveId / tilesN;
    int tn = waveId - tm * tilesN;

    int lane = threadIdx.x & 31;
    int half = lane >> 4;                             // 0 or 1
    int mrow = lane & 15;
    int n0 = tn * 64;
    int m0 = tm * 32;

    // A source rows for the 2 M sub-tiles (clamped; duplicates discarded by
    // the store guard)
    const float* __restrict__ xrow0;
    const float* __restrict__ xrow1;
    {
        int ma = m0 + mrow;
        int mb = m0 + 16 + mrow;
        if (ma >= nrows) ma = nrows - 1;
        if (mb >= nrows) mb = nrows - 1;
        xrow0 = X + (size_t)ma * (size_t)K;
        xrow1 = X + (size_t)mb * (size_t)K;
    }

    v8f acc[8];
    #pragma unroll
    for (int i = 0; i < 8; ++i) acc[i] = (v8f){};

    for (int k0 = 0; k0 < K; k0 += 4) {
        int ka = k0 + 2 * half;                       // K pair owned by this half

        // A fragments: two consecutive K values per row (8B aligned b64 loads)
        v2f a0 = *reinterpret_cast<const v2f*>(xrow0 + ka);
        v2f a1 = *reinterpret_cast<const v2f*>(xrow1 + ka);

        // B fragments: rows ka and ka+1 of W, 4 column sub-tiles of 16
        const float* __restrict__ w0 = W + (size_t)ka * HID + n0 + mrow;
        const float* __restrict__ w1 = w0 + HID;
        v2f b[4];
        #pragma unroll
        for (int j = 0; j < 4; ++j) {
            b[j].x = w0[j * 16];
            b[j].y = w1[j * 16];
        }

        #pragma unroll
        for (int j = 0; j < 4; ++j) {
            acc[j] = __builtin_amdgcn_wmma_f32_16x16x4_f32(
                false, a0, false, b[j], (short)0, acc[j], false, false);
            acc[4 + j] = __builtin_amdgcn_wmma_f32_16x16x4_f32(
                false, a1, false, b[j], (short)0, acc[4 + j], false, false);
        }
    }

    // Store D: row M = m0 + i*16 + r + 8*half, col = n0 + j*16 + mrow
    #pragma unroll
    for (int i = 0; i < 2; ++i) {
        #pragma unroll
        for (int r = 0; r < 8; ++r) {
            int mr = m0 + i * 16 + r + 8 * half;
            if (mr < nrows) {
                float* __restrict__ hrow = H + (size_t)mr * HID + n0 + mrow;
                hrow[0]  = acc[i * 4 + 0][r];
                hrow[16] = acc[i * 4 + 1][r];
                hrow[32] = acc[i * 4 + 2][r];
                hrow[48] = acc[i * 4 + 3][r];
            }
        }
    }
}

// ---------------------------------------------------------------------------
// Aggregation: agg[n][:] = h[n][:] * dinv[n]^2  (self loop), then edge scatter
// float4-vectorized streaming.
// ---------------------------------------------------------------------------
__global__ void agg_init_kernel(const float4* __restrict__ h4,
                                const float* __restrict__ dinv,
                                float4* __restrict__ agg4, int total4) {
    int i = blockIdx.x * blockDim.x + threadIdx.x;
    if (i < total4) {
        int node = i >> 6;                            // (i*4) / HID
        float di = dinv[node];
        float s = di * di;
        float4 v = h4[i];
        v.x *= s; v.y *= s; v.z *= s; v.w *= s;
        agg4[i] = v;
    }
}

// one block (64 threads) per edge; thread t handles channels 4t..4t+3.
// Edge scalars are block-uniform -> scalar loads.
__global__ void agg_edges_kernel(const long long* __restrict__ src,
                                 const long long* __restrict__ dst,
                                 const float* __restrict__ h,
                                 const float* __restrict__ dinv,
                                 float* __restrict__ agg, int e) {
    int ed = blockIdx.x;
    if (ed >= e) return;
    int s = (int)src[ed];
    int d = (int)dst[ed];
    float norm = dinv[s] * dinv[d];
    int c = threadIdx.x << 2;                         // 0,4,...,252
    float4 v = *reinterpret_cast<const float4*>(h + (size_t)s * HID + c);
    float* __restrict__ base = agg + (size_t)d * HID + c;
    atomicAdd(base + 0, v.x * norm);
    atomicAdd(base + 1, v.y * norm);
    atomicAdd(base + 2, v.z * norm);
    atomicAdd(base + 3, v.w * norm);
}

__global__ void bias_relu_kernel(float4* __restrict__ agg4,
                                 const float4* __restrict__ b4, int total4) {
    int i = blockIdx.x * blockDim.x + threadIdx.x;
    if (i < total4) {
        float4 v = agg4[i];
        float4 bb = b4[i & 63];                       // HID/4 = 64 float4s
        v.x += bb.x; v.y += bb.y; v.z += bb.z; v.w += bb.w;
        v.x = v.x > 0.0f ? v.x : 0.0f;
        v.y = v.y > 0.0f ? v.y : 0.0f;
        v.z = v.z > 0.0f ? v.z : 0.0f;
        v.w = v.w > 0.0f ? v.w : 0.0f;
        agg4[i] = v;
    }
}

// ---------------------------------------------------------------------------
// Readout: column sums -> mean -> sigmoid(g @ Wfc + bfc)
// ---------------------------------------------------------------------------
__global__ void zero_kernel(float* __restrict__ p, int n) {
    int i = blockIdx.x * blockDim.x + threadIdx.x;
    if (i < n) p[i] = 0.0f;
}

__global__ void col_sum_kernel(const float* __restrict__ h,
                               float* __restrict__ gsum, int n) {
    int c = threadIdx.x;                              // 0..255
    int start = blockIdx.x * 256;
    int end = start + 256;
    if (end > n) end = n;
    float s = 0.0f;
    for (int node = start; node < end; ++node)
        s += h[(size_t)node * HID + c];
    atomicAdd(&gsum[c], s);
}

__global__ void final_head_kernel(const float* __restrict__ gsum,
                                  const float* __restrict__ Wfc,
                                  const float* __restrict__ bfc,
                                  float* __restrict__ out, float invN) {
    __shared__ float red[256];
    int c = threadIdx.x;
    red[c] = gsum[c] * invN * Wfc[c];                 // Wfc is [HID,1]
    __syncthreads();
    for (int s = 128; s > 0; s >>= 1) {
        if (c < s) red[c] += red[c + s];
        __syncthreads();
    }
    if (c == 0) {
        float z = red[0] + bfc[0];
        out[0] = 1.0f / (1.0f + __expf(-z));
    }
}

// ---------------------------------------------------------------------------
// Launch
// ---------------------------------------------------------------------------
extern "C" void kernel_launch(void* const* d_in, const int* in_sizes, int n_in,
                              void* d_out, int out_size, void* d_ws, size_t ws_size,
                              hipStream_t stream) {
    const float*     x    = (const float*)d_in[0];
    const long long* ei   = (const long long*)d_in[1];   // [2, E] int64
    const float*     W1   = (const float*)d_in[2];
    const float*     b1   = (const float*)d_in[3];
    const float*     W2   = (const float*)d_in[4];
    const float*     b2   = (const float*)d_in[5];
    const float*     Wfc  = (const float*)d_in[6];
    const float*     bfc  = (const float*)d_in[7];
    float*           out  = (float*)d_out;

    const int N = in_sizes[0] / IN_DIM;                  // 50000
    const int E = in_sizes[1] / 2;                       // 800000
    const long long* src = ei;
    const long long* dst = ei + E;

    // Workspace layout (floats)
    float* ws   = (float*)d_ws;
    size_t off  = 0;
    float* dinv = ws + off; off += (size_t)((N + 255) & ~255);
    float* bufA = ws + off; off += (size_t)N * HID;      // h (GEMM output)
    float* bufB = ws + off; off += (size_t)N * HID;      // aggregated / relu'd
    float* gsum = ws + off; off += HID;
    (void)ws_size; (void)n_in; (void)out_size;

    const int T = 256;
    const int total4 = N * HID / 4;

    // 1. symmetric normalization coefficients
    deg_init_kernel<<<(N + T - 1) / T, T, 0, stream>>>(dinv, N);
    deg_accum_kernel<<<(E + T - 1) / T, T, 0, stream>>>(dst, dinv, E);
    deg_to_dinv_kernel<<<(N + T - 1) / T, T, 0, stream>>>(dinv, N);

    int gemmTiles = ((N + 31) / 32) * (HID / 64);
    int gemmBlocks = (gemmTiles + (T / 32) - 1) / (T / 32);

    // 2. layer 1: h1 = x @ W1 ; agg ; bias+relu
    gemm_wmma_f32_kernel<<<gemmBlocks, T, 0, stream>>>(x, W1, bufA, N, IN_DIM);
    agg_init_kernel<<<(total4 + T - 1) / T, T, 0, stream>>>(
        (const float4*)bufA, dinv, (float4*)bufB, total4);
    agg_edges_kernel<<<E, 64, 0, stream>>>(src, dst, bufA, dinv, bufB, E);
    bias_relu_kernel<<<(total4 + T - 1) / T, T, 0, stream>>>(
        (float4*)bufB, (const float4*)b1, total4);

    // 3. layer 2: h2 = h1' @ W2 ; agg ; bias+relu   (bufB -> bufA -> bufB)
    gemm_wmma_f32_kernel<<<gemmBlocks, T, 0, stream>>>(bufB, W2, bufA, N, HID);
    agg_init_kernel<<<(total4 + T - 1) / T, T, 0, stream>>>(
        (const float4*)bufA, dinv, (float4*)bufB, total4);
    agg_edges_kernel<<<E, 64, 0, stream>>>(src, dst, bufA, dinv, bufB, E);
    bias_relu_kernel<<<(total4 + T - 1) / T, T, 0, stream>>>(
        (float4*)bufB, (const float4*)b2, total4);

    // 4. readout + head
    zero_kernel<<<1, T, 0, stream>>>(gsum, HID);
    col_sum_kernel<<<(N + 255) / 256, T, 0, stream>>>(bufB, gsum, N);
    final_head_kernel<<<1, T, 0, stream>>>(gsum, Wfc, bfc, out, 1.0f / (float)N);
}